// IlluDecompLayer_48885317763442
// MI455X (gfx1250) — compile-verified
//
#include <hip/hip_runtime.h>
#include <hip/hip_bf16.h>

typedef float v2f __attribute__((ext_vector_type(2)));
typedef float v8f __attribute__((ext_vector_type(8)));

#define HW    (256 * 256)
#define NBLK  32    // blocks per batch image
#define NWAVE 8     // wave32s per block
#define NITER 8     // iterations per wave: NBLK*NWAVE*NITER*32 == 65536

__device__ __forceinline__ float gamma_pow(float x) {
    // x^2.2 = exp2(2.2*log2 x); handles x==0 -> -inf -> 0 correctly
    return __builtin_amdgcn_exp2f(2.2f * __builtin_amdgcn_logf(x));
}

__global__ __launch_bounds__(256) void k_zero(float* __restrict__ acc) {
    acc[blockIdx.x * 256 + threadIdx.x] = 0.0f;
}

// Streaming pass: per batch, accumulate S = sum_p m_p m_p^T  (m = [SH basis(9) | shading rgb(3) | 0 pad])
// S stored as full 16x16 per batch in acc[b*256 + ...] (WMMA C/D flat layout M*16+N).
__global__ __launch_bounds__(256) void k_accum(
    const float* __restrict__ inputs, const float* __restrict__ masks,
    const float* __restrict__ am,     const float* __restrict__ nm,
    const float* __restrict__ sm,     float* __restrict__ acc)
{
    constexpr float C1 = 0.429043f, C2 = 0.511664f, C3 = 0.743125f,
                    C4 = 0.886227f, C5 = 0.247708f;

    __shared__ float tile[NWAVE][32][16]; // per-wave feature tile (16 KB)
    __shared__ float red[256];            // block-level Gram reduction

    const int b    = blockIdx.y;
    const int tid  = threadIdx.x;
    const int w    = tid >> 5;
    const int lane = tid & 31;
    const int f    = lane & 15;  // feature index for WMMA operand
    const int hv   = lane >> 4;  // lane half selects pixel pair (K half)

    red[tid] = 0.0f;

    v8f Cacc = {0.f, 0.f, 0.f, 0.f, 0.f, 0.f, 0.f, 0.f};
    const long long base3 = (long long)b * HW * 3;

    for (int it = 0; it < NITER; ++it) {
        const int p = ((blockIdx.x * NWAVE + w) * NITER + it) * 32 + lane;
        const long long i3 = base3 + (long long)p * 3;

        const float nx = nm[i3 + 0], ny = nm[i3 + 1], nz = nm[i3 + 2];
        float ft[16];
        ft[0] = C4;
        ft[1] = 2.f * C2 * ny;
        ft[2] = 2.f * C2 * nz;
        ft[3] = 2.f * C2 * nx;
        ft[4] = 2.f * C1 * nx * ny;
        ft[5] = 2.f * C1 * ny * nz;
        ft[6] = C3 * nz * nz - C5;
        ft[7] = 2.f * C1 * nz * nx;
        ft[8] = C1 * (nx * nx - ny * ny);
        #pragma unroll
        for (int c = 0; c < 3; ++c) {
            float x = inputs[i3 + c];
            // divisor am*sm is bounded in [0.01, 0.9801]; v_rcp_f32 (1 ulp)
            // followed by clip(.,0,1)+eps makes the approx error negligible.
            float s = gamma_pow(x) * __builtin_amdgcn_rcpf(am[i3 + c] * sm[i3 + c]);
            s = fminf(fmaxf(s, 0.f), 1.f) + 1e-4f;
            ft[9 + c] = s * masks[i3 + c];
        }
        ft[12] = ft[13] = ft[14] = ft[15] = 0.f;

        if (it + 1 < NITER) { // prefetch next iteration's stream
            const long long n3 = i3 + 32 * 3;
            __builtin_prefetch(&nm[n3], 0, 0);
            __builtin_prefetch(&inputs[n3], 0, 0);
            __builtin_prefetch(&masks[n3], 0, 0);
            __builtin_prefetch(&am[n3], 0, 0);
            __builtin_prefetch(&sm[n3], 0, 0);
        }

        float4* dst = (float4*)&tile[w][lane][0];
        dst[0] = make_float4(ft[0],  ft[1],  ft[2],  ft[3]);
        dst[1] = make_float4(ft[4],  ft[5],  ft[6],  ft[7]);
        dst[2] = make_float4(ft[8],  ft[9],  ft[10], ft[11]);
        dst[3] = make_float4(ft[12], ft[13], ft[14], ft[15]);
        __syncthreads();

        // 8 rank-4 Gram updates: D += M^T(16x4) * M(4x16); A and B operands
        // share the identical per-lane layout, so pass the same registers.
        #pragma unroll
        for (int k = 0; k < 8; ++k) {
            const int p0 = 4 * k + 2 * hv;
            v2f a;
            a.x = tile[w][p0    ][f];
            a.y = tile[w][p0 + 1][f];
            Cacc = __builtin_amdgcn_wmma_f32_16x16x4_f32(
                false, a, false, a, (short)0, Cacc, false, false);
        }
        __syncthreads();
    }

    // C/D layout: VGPR r, lane L -> (M = r + 8*(L>>4), N = L&15)
    #pragma unroll
    for (int r = 0; r < 8; ++r)
        atomicAdd(&red[(r + 8 * hv) * 16 + f], Cacc[r]);
    __syncthreads();
    atomicAdd(&acc[b * 256 + tid], red[tid]);
}

// Tiny solve: x = (A^T A)^{-1} A^T s via 9x9 Cholesky, then PCA via
// pinv(V) = V^T (V V^T)^{-1} with a 10x10 Cholesky. One thread per batch.
__global__ void k_solve(const float* __restrict__ acc,
                        const float* __restrict__ vectors,
                        const float* __restrict__ means,
                        float* __restrict__ out, int B)
{
    const int b = threadIdx.x;
    if (b >= B) return;
    const float* S = acc + b * 256;

    float L[9][9];
    for (int i = 0; i < 9; ++i)
        for (int j = 0; j <= i; ++j) {
            float s = S[i * 16 + j];
            for (int k = 0; k < j; ++k) s -= L[i][k] * L[j][k];
            L[i][j] = (i == j) ? sqrtf(fmaxf(s, 1e-30f)) : s / L[j][j];
        }

    float light[27];
    for (int c = 0; c < 3; ++c) {
        float y[9], x[9];
        for (int i = 0; i < 9; ++i) {
            float s = S[i * 16 + 9 + c];
            for (int k = 0; k < i; ++k) s -= L[i][k] * y[k];
            y[i] = s / L[i][i];
        }
        for (int i = 8; i >= 0; --i) {
            float s = y[i];
            for (int k = i + 1; k < 9; ++k) s -= L[k][i] * x[k];
            x[i] = s / L[i][i];
        }
        for (int i = 0; i < 9; ++i) light[i * 3 + c] = x[i];
    }

    float lm[27];
    for (int j = 0; j < 27; ++j) lm[j] = light[j] - means[j];

    float G[10][10];
    for (int a = 0; a < 10; ++a)
        for (int c = 0; c <= a; ++c) {
            float s = 0.f;
            for (int j = 0; j < 27; ++j) s += vectors[a * 27 + j] * vectors[c * 27 + j];
            G[a][c] = s; G[c][a] = s;
        }
    float z[10];
    for (int k = 0; k < 10; ++k) {
        float s = 0.f;
        for (int j = 0; j < 27; ++j) s += vectors[k * 27 + j] * lm[j];
        z[k] = s;
    }
    float LG[10][10];
    for (int i = 0; i < 10; ++i)
        for (int j = 0; j <= i; ++j) {
            float s = G[i][j];
            for (int k = 0; k < j; ++k) s -= LG[i][k] * LG[j][k];
            LG[i][j] = (i == j) ? sqrtf(fmaxf(s, 1e-30f)) : s / LG[j][j];
        }
    float y10[10], p10[10];
    for (int i = 0; i < 10; ++i) {
        float s = z[i];
        for (int k = 0; k < i; ++k) s -= LG[i][k] * y10[k];
        y10[i] = s / LG[i][i];
    }
    for (int i = 9; i >= 0; --i) {
        float s = y10[i];
        for (int k = i + 1; k < 10; ++k) s -= LG[k][i] * p10[k];
        p10[i] = s / LG[i][i];
    }

    for (int j = 0; j < 27; ++j) {
        float s = means[j];
        for (int k = 0; k < 10; ++k) s += p10[k] * vectors[k * 27 + j];
        out[b * 27 + j] = s;
    }
}

extern "C" void kernel_launch(void* const* d_in, const int* in_sizes, int n_in,
                              void* d_out, int out_size, void* d_ws, size_t ws_size,
                              hipStream_t stream) {
    const float* inputs  = (const float*)d_in[0];
    const float* masks   = (const float*)d_in[1];
    const float* am      = (const float*)d_in[2];
    const float* nm      = (const float*)d_in[3];
    const float* sm      = (const float*)d_in[4];
    const float* vectors = (const float*)d_in[5];
    const float* means   = (const float*)d_in[6];
    float* out = (float*)d_out;
    float* acc = (float*)d_ws; // B*256 floats

    const int B = in_sizes[0] / (HW * 3);

    k_zero<<<dim3(B), dim3(256), 0, stream>>>(acc);
    k_accum<<<dim3(NBLK, B), dim3(256), 0, stream>>>(inputs, masks, am, nm, sm, acc);
    k_solve<<<dim3(1), dim3(32), 0, stream>>>(acc, vectors, means, out, B);
}